// Model_PDE_49031346651791
// MI455X (gfx1250) — compile-verified
//
#include <hip/hip_runtime.h>

typedef __attribute__((ext_vector_type(2))) float v2f;
typedef __attribute__((ext_vector_type(4))) float v4f;
typedef __attribute__((ext_vector_type(8))) float v8f;

#define BATCH 524288
#define NGROUP (BATCH / 16)   // 16 samples per wave-group

// add the partner value from lane^16 (wave32 SWAPX16 swizzle: xor=0x10, and=0x1f)
__device__ __forceinline__ float xsum16(float v) {
    return v + __int_as_float(__builtin_amdgcn_ds_swizzle(__float_as_int(v), 0x401F));
}

__global__ __launch_bounds__(256) void pinn_pde_kernel(
    const float* __restrict__ x,  const float* __restrict__ W1, const float* __restrict__ b1,
    const float* __restrict__ W2, const float* __restrict__ b2,
    const float* __restrict__ wo, const float* __restrict__ bo,
    float* __restrict__ out)
{
    // A-operands for V_WMMA_F32_16X16X4_F32 (per tile t: 32 lanes x float2),
    // K=0,1 carry W[:,0],W[:,1] (lanes 0-15), K=2 carries bias (lanes 16-31), K=3 = 0.
    __shared__ v2f  sA1[128];
    __shared__ v2f  sA2[128];
    __shared__ v4f  sPK[64];    // (wo, W1[:,0], W1[:,1], W2[:,0]) zero-padded to 64
    __shared__ float sW21[64];  // W2[:,1]

    const int tid = threadIdx.x;
    if (tid < 128) {
        int lane = tid & 31, t = tid >> 5;
        int m = (lane & 15) + 16 * t;
        bool valid = m < 50;
        v2f a1, a2;
        if (lane < 16) {            // rows M=0..15 of tile: K0,K1 = weight columns
            a1.x = valid ? W1[2*m]   : 0.0f;  a1.y = valid ? W1[2*m+1] : 0.0f;
            a2.x = valid ? W2[2*m]   : 0.0f;  a2.y = valid ? W2[2*m+1] : 0.0f;
        } else {                    // K2 = bias, K3 = 0
            a1.x = valid ? b1[m] : 0.0f;      a1.y = 0.0f;
            a2.x = valid ? b2[m] : 0.0f;      a2.y = 0.0f;
        }
        sA1[tid] = a1; sA2[tid] = a2;
    }
    if (tid < 64) {
        bool valid = tid < 50;
        v4f p;
        p.x = valid ? wo[tid]      : 0.0f;
        p.y = valid ? W1[2*tid]    : 0.0f;
        p.z = valid ? W1[2*tid+1]  : 0.0f;
        p.w = valid ? W2[2*tid]    : 0.0f;
        sPK[tid]  = p;
        sW21[tid] = valid ? W2[2*tid+1] : 0.0f;
    }
    __syncthreads();

    const int   lane = tid & 31;
    const int   sm   = lane & 15;          // sample within group
    const bool  lo   = lane < 16;
    const int   hb   = (lane >> 4) * 8;    // hidden-row offset for this lane half
    const float bo0  = bo[0];
    const float L2E  = 1.4426950408889634f;

    v2f a1r[4], a2r[4];
    #pragma unroll
    for (int t = 0; t < 4; ++t) { a1r[t] = sA1[t*32 + lane]; a2r[t] = sA2[t*32 + lane]; }

    const int gw      = blockIdx.x * (blockDim.x >> 5) + (tid >> 5);
    const int gstride = gridDim.x  * (blockDim.x >> 5);

    for (int g = gw; g < NGROUP; g += gstride) {
        const int sample = g * 16 + sm;

        // B operand: lanes 0-15 carry (x0, x1) of sample N=lane; lanes 16-31 carry (1, 0)
        v2f bb;
        if (lo) bb = ((const v2f*)x)[sample];
        else    { bb.x = 1.0f; bb.y = 0.0f; }

        float accU = 0.0f, accX0 = 0.0f, accX1 = 0.0f, accXX = 0.0f;

        #pragma unroll
        for (int t = 0; t < 4; ++t) {
            v8f c0 = {};
            // z[h][sample] = W[h,:].x + b[h],  h = r + 8*(lane>=16) + 16*t, sample = lane&15
            v8f z1 = __builtin_amdgcn_wmma_f32_16x16x4_f32(false, a1r[t], false, bb,
                                                           (short)0, c0, false, false);
            v8f z2 = __builtin_amdgcn_wmma_f32_16x16x4_f32(false, a2r[t], false, bb,
                                                           (short)0, c0, false, false);
            #pragma unroll
            for (int r = 0; r < 8; ++r) {
                const int h   = hb + 16*t + r;
                const v4f  ck  = sPK[h];
                const float w21 = sW21[h];
                const float woj = ck.x, w10 = ck.y, w11 = ck.z, w20 = ck.w;

                const float zz1 = z1[r], zz2 = z2[r];
                // sigmoid / tanh via native v_exp_f32 + v_rcp_f32
                const float S = __builtin_amdgcn_rcpf(1.0f + __builtin_amdgcn_exp2f(-zz1 * L2E));
                const float E = __builtin_amdgcn_exp2f(zz2 * (2.0f * L2E));
                const float T = 1.0f - 2.0f * __builtin_amdgcn_rcpf(E + 1.0f);

                const float SP  = S * (1.0f - S);
                const float SPP = SP * (1.0f - 2.0f * S);
                const float TP  = 1.0f - T * T;
                const float TPP = -2.0f * T * TP;
                const float A1  = SP * T;   // multiplies W1[j,k]
                const float A2  = S * TP;   // multiplies W2[j,k]

                accU  = fmaf(woj, S * T, accU);
                accX0 = fmaf(woj, fmaf(A1, w10, A2 * w20), accX0);
                accX1 = fmaf(woj, fmaf(A1, w11, A2 * w21), accX1);
                const float hxx = fmaf(SPP * T, w11 * w11,
                                  fmaf(2.0f * SP * TP, w11 * w21,
                                       S * TPP * (w21 * w21)));
                accXX = fmaf(woj, hxx, accXX);
            }
        }

        // fold the two lane halves (hidden 0-7+16t  vs  8-15+16t)
        accU  = xsum16(accU);
        accX0 = xsum16(accX0);
        accX1 = xsum16(accX1);
        accXX = xsum16(accXX);
        const float xv = xsum16(lo ? bb.y : 0.0f);   // broadcast x1 to hi lanes

        const float uOut  = accU + bo0;
        const float alpha = fmaf(-0.5f, xv, 0.1f);
        // B_F=0.5, SIGMA=1, B_COEF=0.5, C_COEF=0.5, C_F=0.9
        const float pde = 0.5f * xv * xv + accX0 + 0.5f * accXX
                        + 0.5f * xv * accX1 + 0.5f * alpha * accX1
                        + 0.9f * alpha * alpha;

        // lo lanes store u, hi lanes store pde — one coalesced store each
        const int off  = lo ? sample : (BATCH + sample);
        out[off] = lo ? uOut : pde;
    }
}

extern "C" void kernel_launch(void* const* d_in, const int* in_sizes, int n_in,
                              void* d_out, int out_size, void* d_ws, size_t ws_size,
                              hipStream_t stream) {
    const float* x  = (const float*)d_in[0];
    const float* W1 = (const float*)d_in[1];
    const float* b1 = (const float*)d_in[2];
    const float* W2 = (const float*)d_in[3];
    const float* b2 = (const float*)d_in[4];
    const float* wo = (const float*)d_in[5];
    const float* bo = (const float*)d_in[6];
    float* out = (float*)d_out;

    dim3 grid(1024), block(256);  // 8192 waves, 4 sample-groups each (NGROUP = 32768)
    hipLaunchKernelGGL(pinn_pde_kernel, grid, block, 0, stream,
                       x, W1, b1, W2, b2, wo, bo, out);
}